// MindieCif_71244917506095
// MI455X (gfx1250) — compile-verified
//
#include <hip/hip_runtime.h>
#include <hip/hip_bf16.h>

typedef float v2f __attribute__((ext_vector_type(2)));
typedef float v8f __attribute__((ext_vector_type(8)));

#define THRESH 1.0f
#define NT 4   // N-tiles (16 cols each) per wave in the WMMA kernel

// ---------------------------------------------------------------------------
// Kernel 1: per-batch sequential CIF scalar scan (lane b = batch b, one wave).
// Alphas staged through LDS in 32-wide tiles. For b=0 emits a packed per-time
// record wpack[t] = {wcur, wrem(at fires only), seg-as-int-bits, 0} so the
// WMMA kernel's A-side is a single b128 load per timestep:
//   W[j][t] = wcur[t]*(seg[t]==j) + wrem[t]*(seg[t]==j-1)
// Also: fire times (b=0), per-batch tail info, final integrate.
// ---------------------------------------------------------------------------
__global__ __launch_bounds__(32) void cif_scan_kernel(
    const float* __restrict__ alphas,     // [B,T]
    const float* __restrict__ integrate,  // [B]
    float4* __restrict__ wpack,           // [T]
    int*   __restrict__ firet,            // [T] time of j-th fire (b=0)
    int*   __restrict__ nf_ptr,           // #fires (b=0)
    int*   __restrict__ t_last,           // [B]
    float* __restrict__ rem_last,         // [B]
    float* __restrict__ out_integ,        // [B] final integrate
    int B, int T)
{
    __shared__ float atile[32][33];
    const int lane = threadIdx.x;
    const int b = lane;
    float integ = (b < B) ? integrate[b] : 0.0f;
    int   nf = 0;
    int   lastt = -1;
    float lastrem = 0.0f;

    for (int tt = 0; tt < T; tt += 32) {
        for (int bb = 0; bb < B; ++bb) {
            int t = tt + lane;
            atile[bb][lane] = (t < T) ? alphas[(size_t)bb * T + t] : 0.0f;
        }
        __syncthreads();
        if (b < B) {
            int kend = T - tt; if (kend > 32) kend = 32;
            for (int k = 0; k < kend; ++k) {
                int t = tt + k;
                float alpha = atile[b][k];
                float dist  = 1.0f - integ;          // dist_completion (pre-add)
                float post  = integ + alpha;         // emitted "fires" value
                bool  fire  = (post >= THRESH);
                float cur   = fire ? dist : alpha;   // weight spent this step
                float rem   = alpha - cur;           // leftover after firing
                if (b == 0) {
                    wpack[t] = make_float4(cur, fire ? rem : 0.0f,
                                           __int_as_float(nf), 0.0f);
                    if (fire) firet[nf] = t;
                }
                if (fire) { lastt = t; lastrem = rem; ++nf; integ = post - 1.0f; }
                else      { integ = post; }
            }
        }
        __syncthreads();
    }
    if (b == 0) *nf_ptr = nf;
    if (b < B) {
        t_last[b]   = lastt;
        rem_last[b] = lastrem;
        out_integ[b] = integ;
    }
}

// ---------------------------------------------------------------------------
// Kernel 2: fired frames via fp32 WMMA. One wave per (16 fire rows x 64 H
// cols). For each K-chunk of 4 timesteps: one shared A fragment (banded
// segment weights, built branch-free from a single float4 stream) feeds 4
// independent V_WMMA_F32_16X16X4_F32 accumulators (4 N-tiles).
// Fragment layouts per CDNA5 ISA 7.12.2:
//   A 16x4 f32: lane M=lane&15, vgprs hold K=(lane>>4)*2+{0,1}
//   B 4x16 f32: lane N=lane&15, vgprs hold K=(lane>>4)*2+{0,1}
//   C/D 16x16 f32: vgpr v -> row v + 8*(lane>>4), col = lane&15
// All in-loop selection is register ternaries (v_cndmask): EXEC stays all-1s.
// ---------------------------------------------------------------------------
__global__ __launch_bounds__(32) void cif_fire_wmma_kernel(
    const float*  __restrict__ hidden0,   // hidden batch 0: [T,H]
    const float*  __restrict__ frame0,    // initial frame row 0: [H]
    const float4* __restrict__ wpack,     // [T]
    const int*    __restrict__ firet,
    const int*    __restrict__ nf_ptr,
    float* __restrict__ out_sel,          // [T,H]
    int T, int H)
{
    const int j0 = blockIdx.x * 16;            // fire-row tile
    const int h0 = blockIdx.y * (16 * NT);     // H tile (64 cols)
    const int nf = *nf_ptr;
    if (j0 >= nf) return;                      // uniform: pad rows filled later

    const int lane = threadIdx.x;
    const int half = lane >> 4;                // K-pair select
    const int n    = lane & 15;
    const int jm   = j0 + (lane & 15);         // A row

    const int tstart = (j0 == 0) ? 0 : firet[j0 - 1];
    int jlast = j0 + 15; if (jlast > nf - 1) jlast = nf - 1;
    const int tend = firet[jlast];
    const int tmax = T - 1;

    // per-u clamped column offsets (clamped cols never stored)
    int col[NT];
#pragma unroll
    for (int u = 0; u < NT; ++u) {
        int cu = h0 + u * 16 + n;
        col[u] = cu < (H - 1) ? cu : (H - 1);
    }

    v8f c[NT];
#pragma unroll
    for (int u = 0; u < NT; ++u)
#pragma unroll
        for (int v = 0; v < 8; ++v) c[u][v] = 0.0f;
    if (j0 == 0) {   // segment 0 carries the initial frame (row M=0 = vgpr0)
#pragma unroll
        for (int u = 0; u < NT; ++u) {
            float f0 = frame0[col[u]];
            c[u][0] = (half == 0) ? f0 : 0.0f;
        }
    }

    for (int tk = (tstart & ~3); tk <= tend; tk += 4) {
        int ta = tk + half * 2;
        int tb = ta + 1;
        int tca = ta < tmax ? ta : tmax;       // clamped addresses; weights
        int tcb = tb < tmax ? tb : tmax;       // forced to 0 when t > tend
        float4 wa = wpack[tca];
        float4 wb = wpack[tcb];
        int sa = __float_as_int(wa.z);
        int sb = __float_as_int(wb.z);
        float ax = (sa == jm) ? wa.x : ((sa == jm - 1) ? wa.y : 0.0f);
        float ay = (sb == jm) ? wb.x : ((sb == jm - 1) ? wb.y : 0.0f);
        ax = (ta <= tend) ? ax : 0.0f;
        ay = (tb <= tend) ? ay : 0.0f;
        v2f a; a.x = ax; a.y = ay;

        int ra = tca * H;                      // 32-bit element offsets
        int rb = tcb * H;
        v2f bf[NT];
#pragma unroll
        for (int u = 0; u < NT; ++u) {
            bf[u].x = hidden0[ra + col[u]];
            bf[u].y = hidden0[rb + col[u]];
        }
#pragma unroll
        for (int u = 0; u < NT; ++u)
            c[u] = __builtin_amdgcn_wmma_f32_16x16x4_f32(
                       false, a, false, bf[u], (short)0, c[u], false, false);
    }

#pragma unroll
    for (int u = 0; u < NT; ++u) {
        int cu = h0 + u * 16 + n;
        if (cu < H) {
#pragma unroll
            for (int v = 0; v < 8; ++v) {
                int row = j0 + v + half * 8;
                out_sel[row * H + cu] = c[u][v];
            }
        }
    }
}

// ---------------------------------------------------------------------------
// Kernel 3: pad rows. idx fill_value=0 => frames[0][0] = frame_init + cur[0]*h[0,0,:]
// Overwrites all rows >= nf (runs after the WMMA kernel in stream order).
// ---------------------------------------------------------------------------
__global__ void cif_padfill_kernel(
    const float*  __restrict__ hidden0,   // [T,H] (row t=0 used)
    const float*  __restrict__ frame0,    // [H]
    const float4* __restrict__ wpack,     // wpack[0].x = cur at t=0
    const int*    __restrict__ nf_ptr,
    float* __restrict__ out_sel, int T, int H)
{
    long long idx = (long long)blockIdx.x * blockDim.x + threadIdx.x;
    if (idx >= (long long)T * H) return;
    int row = (int)(idx / H);
    int h   = (int)(idx - (long long)row * H);
    if (row < *nf_ptr) return;
    out_sel[idx] = frame0[h] + wpack[0].x * hidden0[h];
}

// ---------------------------------------------------------------------------
// Kernel 4: final frame per batch = rem_last*h[t_last] + sum_{t>t_last} a*h
// (or frame_init + full sum if that batch never fired).
// ---------------------------------------------------------------------------
__global__ void cif_tail_kernel(
    const float* __restrict__ hidden,     // [B,T,H]
    const float* __restrict__ alphas,     // [B,T]
    const float* __restrict__ frame,      // [B,H]
    const int*   __restrict__ t_last,
    const float* __restrict__ rem_last,
    float* __restrict__ out_fnew,         // [B,H]
    int T, int H)
{
    int b = blockIdx.x;
    int h = blockIdx.y * blockDim.x + threadIdx.x;
    if (h >= H) return;
    int tl = t_last[b];
    const float* hb = hidden + (size_t)b * T * H;
    float acc = (tl >= 0) ? rem_last[b] * hb[(size_t)tl * H + h]
                          : frame[(size_t)b * H + h];
    for (int t = tl + 1; t < T; ++t)
        acc += alphas[(size_t)b * T + t] * hb[(size_t)t * H + h];
    out_fnew[(size_t)b * H + h] = acc;
}

// ---------------------------------------------------------------------------
extern "C" void kernel_launch(void* const* d_in, const int* in_sizes, int n_in,
                              void* d_out, int out_size, void* d_ws, size_t ws_size,
                              hipStream_t stream) {
    const float* hidden    = (const float*)d_in[0];  // [B,T,H]
    const float* alphas    = (const float*)d_in[1];  // [B,T]
    const float* integrate = (const float*)d_in[2];  // [B]
    const float* frame     = (const float*)d_in[3];  // [B,H]

    const int B = in_sizes[2];                 // 32 (scan kernel assumes B<=32)
    const int T = in_sizes[1] / B;             // 2048
    const int H = in_sizes[0] / in_sizes[1];   // 512

    // workspace layout (4B units): wpack[4T] | firet[T] | nf | t_last[B] | rem_last[B]
    float*  wsf      = (float*)d_ws;
    float4* wpack    = (float4*)wsf;
    int*    firet    = (int*)(wsf + 4 * (size_t)T);
    int*    nf_ptr   = (int*)(wsf + 5 * (size_t)T);
    int*    t_lastp  = nf_ptr + 1;
    float*  rem_lastp = (float*)(t_lastp + B);

    // output layout: frame_sel [T*H] | integ_new [B] | frame_new [B*H]
    float* out       = (float*)d_out;
    float* out_sel   = out;
    float* out_integ = out + (size_t)T * H;
    float* out_fnew  = out_integ + B;

    cif_scan_kernel<<<1, 32, 0, stream>>>(alphas, integrate, wpack, firet,
                                          nf_ptr, t_lastp, rem_lastp,
                                          out_integ, B, T);

    dim3 g2((T + 15) / 16, (H + 16 * NT - 1) / (16 * NT));
    cif_fire_wmma_kernel<<<g2, 32, 0, stream>>>(hidden, frame, wpack, firet,
                                                nf_ptr, out_sel, T, H);

    long long tot = (long long)T * H;
    cif_padfill_kernel<<<(unsigned)((tot + 255) / 256), 256, 0, stream>>>(
        hidden, frame, wpack, nf_ptr, out_sel, T, H);

    dim3 g3(B, (H + 255) / 256);
    cif_tail_kernel<<<g3, 256, 0, stream>>>(hidden, alphas, frame, t_lastp,
                                            rem_lastp, out_fnew, T, H);
}